// QConv1d_87050397156032
// MI455X (gfx1250) — compile-verified
//
#include <hip/hip_runtime.h>
#include <hip/hip_bf16.h>

typedef __attribute__((ext_vector_type(2))) float v2f;
typedef __attribute__((ext_vector_type(8))) float v8f;

#define R_TOT   8168
#define LOUT    2042
#define CO      64
#define D_TOT   112
#define NTOT    43911168u   // 3*28*8168*64
#define HALFN   21955584u   // NTOT/2  (== 42*522752)
#define TAB_N   61000       // 61*1000

// ---- JAX Threefry-2x32 (20 rounds), bit-exact with jax._src.prng ----
__host__ __device__ inline void threefry2x32(unsigned ks0, unsigned ks1,
                                             unsigned x0, unsigned x1,
                                             unsigned &o0, unsigned &o1) {
  unsigned ks2 = ks0 ^ ks1 ^ 0x1BD11BDAu;
  x0 += ks0; x1 += ks1;
#define TF_R(r) { x0 += x1; x1 = (x1 << (r)) | (x1 >> (32 - (r))); x1 ^= x0; }
  TF_R(13) TF_R(15) TF_R(26) TF_R(6)   x0 += ks1; x1 += ks2 + 1u;
  TF_R(17) TF_R(29) TF_R(16) TF_R(24)  x0 += ks2; x1 += ks0 + 2u;
  TF_R(13) TF_R(15) TF_R(26) TF_R(6)   x0 += ks0; x1 += ks1 + 3u;
  TF_R(17) TF_R(29) TF_R(16) TF_R(24)  x0 += ks1; x1 += ks2 + 4u;
  TF_R(13) TF_R(15) TF_R(26) TF_R(6)   x0 += ks2; x1 += ks0 + 5u;
#undef TF_R
  o0 = x0; o1 = x1;
}

// x_lin[r, col] in "hardware integer" domain: round(clip(x,-2,1.75)*4)*2 + 6
__device__ inline float load_xint(const float* __restrict__ x, int b, int l, int col) {
  int cin = col / 7;
  int kk  = col - cin * 7;
  float v = x[(b * 16 + cin) * 2048 + l + kk];
  return rintf(fminf(fmaxf(v, -2.f), 1.75f) * 4.f) * 2.f + 6.f;
}

// ---- prep: w3[3][64][112] from weight ----
__global__ void prep_w3_kernel(const float* __restrict__ w, float* __restrict__ w3) {
  int idx = blockIdx.x * blockDim.x + threadIdx.x;
  if (idx >= CO * D_TOT) return;
  int c = idx / D_TOT, d = idx - c * D_TOT;
  int cin = d / 7, k = d - cin * 7;
  float wv = w[(c * 16 + cin) * 7 + k];
  float qw = rintf(fminf(fmaxf(wv, -2.f), 1.75f) * 4.f) * 0.25f;
  float qi = 2.f * qw;
  float qp  = fmaxf(qi, 0.f);
  float qmh = fmaxf(-qi - 1.f, 0.f);
  float qml = fmaxf(-qi, 0.f) - 2.f * qmh;
  w3[0 * CO * D_TOT + idx] = qp;
  w3[1 * CO * D_TOT + idx] = qml;
  w3[2 * CO * D_TOT + idx] = qmh;
}

// ---- prep: bias[64] = 6 * sum(q_int) ----
__global__ void prep_bias_kernel(const float* __restrict__ w, float* __restrict__ bias) {
  int c = threadIdx.x;
  if (c >= CO) return;
  float s = 0.f;
  for (int d = 0; d < D_TOT; ++d) {
    int cin = d / 7, k = d - cin * 7;
    float wv = w[(c * 16 + cin) * 7 + k];
    float qw = rintf(fminf(fmaxf(wv, -2.f), 1.75f) * 4.f) * 0.25f;
    s += 2.f * qw;
  }
  bias[c] = 6.f * s;
}

// ---- main: WMMA 16x16x4 f32 grouped dots + LDS table gather + paired threefry ----
__global__ __launch_bounds__(256) void qconv_main(
    const float* __restrict__ x, const float* __restrict__ table,
    const float* __restrict__ w3, const float* __restrict__ bias,
    float* __restrict__ out,
    unsigned k1a, unsigned k1b, unsigned k2a, unsigned k2b) {
  extern __shared__ float ltab[];   // 61*1000 floats = 244000 B (<320KB LDS/WGP)

  // cooperative 128b-coalesced table load into LDS
  {
    const float4* t4 = reinterpret_cast<const float4*>(table);
    float4* l4 = reinterpret_cast<float4*>(ltab);
    for (int i = threadIdx.x; i < TAB_N / 4; i += blockDim.x) l4[i] = t4[i];
  }
  __syncthreads();

  const int lane = threadIdx.x & 31;
  const int wave = threadIdx.x >> 5;
  const int hi   = lane >> 4;          // lane half: selects K pairs (0,1)/(2,3)
  const int M    = lane & 15;          // A row index for this lane
  const int NN   = lane & 15;          // B/D column index for this lane

  const float s3[3] = {15.f, -15.f, -30.f};  // signs * 15

  const int numJobs = (R_TOT + 15) / 16 * 4;     // 511 row tiles * 4 col tiles
  for (int job = blockIdx.x * 8 + wave; job < numJobs; job += gridDim.x * 8) {
    const int rt = job >> 2, ct = job & 3;
    const int r0 = rt * 16, c0 = ct * 16;
    const int myC = c0 + NN;

    // A-operand row for this lane (clamped; out-of-range rows are computed
    // unguarded -- always memory-safe -- and masked only at the final store)
    const int rA = min(r0 + M, R_TOT - 1);
    const int bA = rA / LOUT, lA = rA - bA * LOUT;
    // per-lane (r*CO + c) base for D-layout element j: r = r0 + j + 8*hi
    const unsigned rcBase = (unsigned)((r0 + 8 * hi) * CO + myC);

    float acc[8];
#pragma unroll
    for (int j = 0; j < 8; ++j) acc[j] = 0.f;

    // 42 counter-pairs: tg1 = p (< HALFN side), tg2 = p + 42.
    // One threefry eval per key yields random bits for BOTH partners.
    for (int p = 0; p < 42; ++p) {
      const int t1 = (p < 28) ? 0 : 1;
      const int g1 = (p < 28) ? p : p - 28;
      const int q  = p + 42;
      const int t2 = (q < 56) ? 1 : 2;
      const int g2 = (q < 56) ? q - 28 : q - 56;

      // --- WMMA #1: y[t1,g1] tile ---
      const int cb1 = 4 * g1 + 2 * hi;
      v2f A1; A1.x = load_xint(x, bA, lA, cb1); A1.y = load_xint(x, bA, lA, cb1 + 1);
      const float* w1 = w3 + (t1 * CO + myC) * D_TOT + cb1;
      v2f B1; B1.x = w1[0]; B1.y = w1[1];
      v8f cz1 = {};
      v8f d1 = __builtin_amdgcn_wmma_f32_16x16x4_f32(
          false, A1, false, B1, (short)0, cz1, false, false);

      // --- WMMA #2: y[t2,g2] tile ---
      const int cb2 = 4 * g2 + 2 * hi;
      v2f A2; A2.x = load_xint(x, bA, lA, cb2); A2.y = load_xint(x, bA, lA, cb2 + 1);
      const float* w2 = w3 + (t2 * CO + myC) * D_TOT + cb2;
      v2f B2; B2.x = w2[0]; B2.y = w2[1];
      v8f cz2 = {};
      v8f d2 = __builtin_amdgcn_wmma_f32_16x16x4_f32(
          false, A2, false, B2, (short)0, cz2, false, false);

      const unsigned fBase = (unsigned)p * 522752u + rcBase;  // p * R_TOT*CO + rc
      const float sgn1 = s3[t1];
      const float sgn2 = s3[t2];

#pragma unroll
      for (int j = 0; j < 8; ++j) {
        const float y1 = d1[j], y2 = d2[j];
        int yi1 = (int)y1; yi1 = yi1 < 0 ? 0 : (yi1 > 60 ? 60 : yi1);
        int yi2 = (int)y2; yi2 = yi2 < 0 ? 0 : (yi2 > 60 ? 60 : yi2);
        const unsigned f = fBase + (unsigned)(j * CO);
        unsigned hb1, hb2, lb1, lb2;
        threefry2x32(k1a, k1b, f, f + HALFN, hb1, hb2);   // higher bits: tg1 & tg2
        threefry2x32(k2a, k2b, f, f + HALFN, lb1, lb2);   // lower  bits: tg1 & tg2
        const unsigned rnd1 = ((hb1 % 1000u) * 296u + (lb1 % 1000u)) % 1000u;
        const unsigned rnd2 = ((hb2 % 1000u) * 296u + (lb2 % 1000u)) % 1000u;
        acc[j] += sgn1 * ltab[yi1 * 1000 + rnd1] + sgn2 * ltab[yi2 * 1000 + rnd2];
      }
    }

    // epilogue: bias, /16, output fake_quant (Qn=2, nb=6 -> clip[-8,7.75], scale 4)
    const float bs = bias[myC];
#pragma unroll
    for (int j = 0; j < 8; ++j) {
      const int r = r0 + j + 8 * hi;
      if (r < R_TOT) {
        const int b = r / LOUT, l = r - b * LOUT;
        float v = (acc[j] - bs) * 0.0625f;
        float qv = rintf(fminf(fmaxf(v, -8.f), 7.75f) * 4.f) * 0.25f;
        out[(b * CO + myC) * LOUT + l] = qv;
      }
    }
  }
}

extern "C" void kernel_launch(void* const* d_in, const int* in_sizes, int n_in,
                              void* d_out, int out_size, void* d_ws, size_t ws_size,
                              hipStream_t stream) {
  const float* x      = (const float*)d_in[0];
  const float* weight = (const float*)d_in[1];
  const float* table  = (const float*)d_in[2];
  float* out  = (float*)d_out;
  float* w3   = (float*)d_ws;                 // 3*64*112 floats
  float* bias = w3 + 3 * CO * D_TOT;          // 64 floats

  // Derive split(key(1234)) on host (deterministic pure integer math):
  // counts iota(4) -> pairs (0,2) and (1,3) under key (0,1234);
  // k1 = (o0(0,2), o0(1,3)), k2 = (o1(0,2), o1(1,3)).
  unsigned e00, e01, e10, e11;
  threefry2x32(0u, 1234u, 0u, 2u, e00, e01);
  threefry2x32(0u, 1234u, 1u, 3u, e10, e11);
  const unsigned k1a = e00, k1b = e10, k2a = e01, k2b = e11;

  prep_w3_kernel  <<<28, 256, 0, stream>>>(weight, w3);
  prep_bias_kernel<<<1, 64, 0, stream>>>(weight, bias);
  qconv_main<<<128, 256, TAB_N * sizeof(float), stream>>>(
      x, table, w3, bias, out, k1a, k1b, k2a, k2b);
}